// BBoxRoIExtractor_37761352466881
// MI455X (gfx1250) — compile-verified
//
#include <hip/hip_runtime.h>
#include <math.h>

// RoIPool (torchvision quantized max pool), C=128, H=W=56 features, 7x7 bins.
// One block (256 threads = 8 wave32) per ROI.
//
// MI455X path: per-ROI bounding rectangle is staged 4 channels at a time into
// LDS with gfx1250 GLOBAL_LOAD_ASYNC_TO_LDS_B128 (ASYNCcnt-tracked DMA, no VGPR
// round trip), double-buffered so chunk k+1's DMA overlaps chunk k's bin-max
// compute out of LDS. 2 x 4 x 56x56 fp32 = 100 KB static LDS (320 KB/WGP on
// CDNA5). Falls back to a direct cache-gather kernel (global_prefetch_b8) if
// the async builtins are unavailable. WMMA is inapplicable: ragged max-gather.

#define OUT_H 7
#define OUT_W 7
#define NBINS (OUT_H * OUT_W)
#define FC 128
#define FH 56
#define FW 56
#define PLANE (FH * FW)
#define CCH 4                       // channels per pipeline chunk
#define NCHUNK (FC / CCH)

#if defined(__has_builtin)
#if __has_builtin(__builtin_amdgcn_global_load_async_to_lds_b128)
#define HAVE_ASYNC_LDS 1
#endif
#endif

__device__ __forceinline__ void compute_bin_tables(
    const float* __restrict__ rois, int roi, int tid,
    int* s_ws, int* s_we, int* s_hs, int* s_he)
{
    if (tid < OUT_W + OUT_H) {
        // cxcywh (normalized) -> pixel xyxy, C-style round = floor(v + 0.5)
        const float cx = rois[roi * 4 + 0] * (float)FW;
        const float cy = rois[roi * 4 + 1] * (float)FH;
        const float rw = rois[roi * 4 + 2] * (float)FW;
        const float rh = rois[roi * 4 + 3] * (float)FH;
        const float x1 = floorf(cx - rw * 0.5f + 0.5f);
        const float y1 = floorf(cy - rh * 0.5f + 0.5f);
        const float x2 = floorf(cx + rw * 0.5f + 0.5f);
        const float y2 = floorf(cy + rh * 0.5f + 0.5f);
        const float bw = fmaxf(x2 - x1 + 1.0f, 1.0f) / (float)OUT_W;
        const float bh = fmaxf(y2 - y1 + 1.0f, 1.0f) / (float)OUT_H;
        if (tid < OUT_W) {
            const float p  = (float)tid;
            const float ws = floorf(p * bw) + x1;
            const float we = ceilf((p + 1.0f) * bw) + x1;
            s_ws[tid] = (int)fminf(fmaxf(ws, 0.0f), (float)FW);
            s_we[tid] = (int)fminf(fmaxf(we, 0.0f), (float)FW);
        } else {
            const int   j  = tid - OUT_W;
            const float p  = (float)j;
            const float hs = floorf(p * bh) + y1;
            const float he = ceilf((p + 1.0f) * bh) + y1;
            s_hs[j] = (int)fminf(fmaxf(hs, 0.0f), (float)FH);
            s_he[j] = (int)fminf(fmaxf(he, 0.0f), (float)FH);
        }
    }
}

// ---------------------------------------------------------------------------
// Fallback: direct cache-gather kernel (L2-resident input, prefetch next rect)
// ---------------------------------------------------------------------------
__global__ __launch_bounds__(256) void roipool_direct_kernel(
    const float* __restrict__ feat, const float* __restrict__ rois,
    float* __restrict__ out, int N)
{
    const int roi = blockIdx.x;
    const int b   = roi / N;
    __shared__ int s_ws[OUT_W], s_we[OUT_W], s_hs[OUT_H], s_he[OUT_H];
    const int tid = threadIdx.x;

    compute_bin_tables(rois, roi, tid, s_ws, s_we, s_hs, s_he);
    __syncthreads();

    const float* __restrict__ fb = feat + (size_t)b   * (FC * PLANE);
    float*       __restrict__ ob = out  + (size_t)roi * (FC * NBINS);

    for (int idx = tid; idx < FC * NBINS; idx += 256) {
        const int c   = idx / NBINS;
        const int bin = idx - c * NBINS;
        const int ph  = bin / OUT_W;
        const int pw  = bin - ph * OUT_W;
        const int hs = s_hs[ph], he = s_he[ph];
        const int ws = s_ws[pw], we = s_we[pw];

        const int nidx = idx + 256;
        if (nidx < FC * NBINS) {
            const int nc   = nidx / NBINS;
            const int nbin = nidx - nc * NBINS;
            const int nph  = nbin / OUT_W;
            const int npw  = nbin - nph * OUT_W;
            __builtin_prefetch(
                fb + (size_t)nc * PLANE + (size_t)s_hs[nph] * FW + s_ws[npw], 0, 0);
        }

        float m = 0.0f;
        if (hs < he && ws < we) {
            m = -3.402823466e+38f;
            const float* __restrict__ fch = fb + (size_t)c * PLANE;
            for (int h = hs; h < he; ++h) {
                const float* __restrict__ row = fch + h * FW;
#pragma unroll 4
                for (int w = ws; w < we; ++w) m = fmaxf(m, row[w]);
            }
        }
        ob[idx] = m;
    }
}

// ---------------------------------------------------------------------------
// Main path: async b128 DMA into LDS, double-buffered over 4-channel chunks
// ---------------------------------------------------------------------------
#if defined(HAVE_ASYNC_LDS)

typedef int v4i_t __attribute__((ext_vector_type(4)));
typedef __attribute__((address_space(1))) v4i_t* gptr_b128;  // global v4i*
typedef __attribute__((address_space(3))) v4i_t* lptr_b128;  // LDS v4i*

__global__ __launch_bounds__(256) void roipool_async_kernel(
    const float* __restrict__ feat, const float* __restrict__ rois,
    float* __restrict__ out, int N)
{
    __shared__ int s_ws[OUT_W], s_we[OUT_W], s_hs[OUT_H], s_he[OUT_H];
    // 2 buffers x 4 channels x (<=56 rows x <=56 cols) fp32 = 100 KB of 320 KB/WGP
    __shared__ __attribute__((aligned(16))) float s_buf[2][CCH * PLANE];

    const int roi = blockIdx.x;
    const int b   = roi / N;
    const int tid = threadIdx.x;

    compute_bin_tables(rois, roi, tid, s_ws, s_we, s_hs, s_he);
    __syncthreads();

    // Union rectangle of all 7x7 bins; columns in 16B (4-float) groups.
    const int r0 = s_hs[0];
    const int r1 = s_he[OUT_H - 1];
    const int rectH = (r1 > r0) ? (r1 - r0) : 0;
    const int g0 = s_ws[0] >> 2;
    const int g1 = (s_we[OUT_W - 1] + 3) >> 2;
    const int nG = (g1 > g0) ? (g1 - g0) : 0;
    const int rowW = nG << 2;               // floats per staged row
    const int segPerCh = rectH * nG;        // b128 segments per channel
    const int nseg = CCH * segPerCh;        // per chunk

    const float* __restrict__ fb = feat + (size_t)b   * (FC * PLANE);
    float*       __restrict__ ob = out  + (size_t)roi * (FC * NBINS);

    // ---- stage chunk 0 into buffer 0 (async b128 -> LDS) ----
    for (int s = tid; s < nseg; s += 256) {
        const int ch  = s / segPerCh;
        const int rem = s - ch * segPerCh;
        const int r   = rem / nG;
        const int g   = rem - r * nG;
        const float* gp = fb + (size_t)ch * PLANE
                        + (size_t)(r0 + r) * FW + ((g0 + g) << 2);
        float* lp = &s_buf[0][(ch * rectH + r) * rowW + (g << 2)];
        __builtin_amdgcn_global_load_async_to_lds_b128(
            (gptr_b128)gp, (lptr_b128)lp, 0, 0);
    }

    for (int k = 0; k < NCHUNK; ++k) {
        // Drain this wave's async DMA for chunk k, then make LDS visible
        // block-wide. (Chunk k+1's DMA, issued below, stays in flight.)
        asm volatile("s_wait_asynccnt 0x0" ::: "memory");
        __syncthreads();

        // ---- issue async DMA for chunk k+1 into the other buffer ----
        if (k + 1 < NCHUNK) {
            const int c0 = (k + 1) * CCH;
            float* buf = s_buf[(k + 1) & 1];
            for (int s = tid; s < nseg; s += 256) {
                const int ch  = s / segPerCh;
                const int rem = s - ch * segPerCh;
                const int r   = rem / nG;
                const int g   = rem - r * nG;
                const float* gp = fb + (size_t)(c0 + ch) * PLANE
                                + (size_t)(r0 + r) * FW + ((g0 + g) << 2);
                float* lp = &buf[(ch * rectH + r) * rowW + (g << 2)];
                __builtin_amdgcn_global_load_async_to_lds_b128(
                    (gptr_b128)gp, (lptr_b128)lp, 0, 0);
            }
        }

        // ---- compute chunk k's CCH x 49 bin maxes out of LDS ----
        if (tid < CCH * NBINS) {
            const float* buf = s_buf[k & 1];
            const int ch  = tid / NBINS;
            const int bin = tid - ch * NBINS;
            const int ph  = bin / OUT_W;
            const int pw  = bin - ph * OUT_W;
            const int hs = s_hs[ph], he = s_he[ph];
            const int ws = s_ws[pw], we = s_we[pw];
            float m = 0.0f;                       // empty bin -> 0
            if (hs < he && ws < we) {
                m = -3.402823466e+38f;            // finfo(float32).min
                for (int h = hs; h < he; ++h) {
                    const float* lrow =
                        buf + (ch * rectH + (h - r0)) * rowW - (g0 << 2);
#pragma unroll 4
                    for (int w = ws; w < we; ++w) m = fmaxf(m, lrow[w]);
                }
            }
            ob[(size_t)(k * CCH + ch) * NBINS + bin] = m;
        }
        // No tail barrier needed: next iteration's wait+barrier separates
        // this chunk's LDS readers from the restage of this buffer (k+2).
    }
}
#endif  // HAVE_ASYNC_LDS

extern "C" void kernel_launch(void* const* d_in, const int* in_sizes, int n_in,
                              void* d_out, int out_size, void* d_ws, size_t ws_size,
                              hipStream_t stream) {
    const float* feat = (const float*)d_in[0];   // (B,128,56,56) fp32
    const float* rois = (const float*)d_in[1];   // (B,N,4) fp32
    float* out = (float*)d_out;                  // (B,N,128,7,7) fp32

    int B = in_sizes[0] / (FC * PLANE);
    if (B < 1) B = 1;
    int N = (in_sizes[1] / 4) / B;
    if (N < 1) N = 1;
    const int nroi = B * N;

#if defined(HAVE_ASYNC_LDS)
    roipool_async_kernel<<<nroi, 256, 0, stream>>>(feat, rois, out, N);
#else
    roipool_direct_kernel<<<nroi, 256, 0, stream>>>(feat, rois, out, N);
#endif
}